// LocalWalk_78640851190128
// MI455X (gfx1250) — compile-verified
//
#include <hip/hip_runtime.h>

typedef __attribute__((ext_vector_type(2))) float v2f;
typedef __attribute__((ext_vector_type(4))) float v4f;
typedef __attribute__((ext_vector_type(8))) float v8f;

#define B_DIM 4
#define C_DIM 128
#define H_DIM 64
#define W_DIM 64
#define HW (H_DIM * W_DIM)
#define RAD 12
#define PWIN 25
#define DH_GROUPS 5
#define DH_PER_GROUP 5
#define EXP_M10 4.5399929762484854e-05f  // expf(-10)

// Pass 1: vectorized zero-fill of the 256MB output (the dominant HBM cost).
// Non-temporal: pure streaming write, keep L2 for the input tensors.
__global__ __launch_bounds__(256) void zero_kernel(v4f* __restrict__ out, long long n4) {
  long long i = (long long)blockIdx.x * blockDim.x + threadIdx.x;
  if (i < n4) {
    v4f z = {0.f, 0.f, 0.f, 0.f};
    __builtin_nontemporal_store(z, out + i);
  }
}

// Pass 2: WMMA local correlation + exp + scatter.
// Grid: (B*H, DH_GROUPS) blocks of 128 threads (4 waves).
// Wave wv handles query tile (b, h, w0..w0+15), w0 = wv*16, for the 5 window
// rows dh in [-12 + 5*blockIdx.y, ...+4].
__global__ __launch_bounds__(128) void corr_kernel(const float* __restrict__ qry,
                                                   const float* __restrict__ key,
                                                   float* __restrict__ out) {
  const int lane = threadIdx.x & 31;
  const int wv   = threadIdx.x >> 5;
  const int b = blockIdx.x / H_DIM;
  const int h = blockIdx.x % H_DIM;
  const int w0 = wv * 16;
  const int half = lane >> 4;   // which half-wave
  const int l15  = lane & 15;
  const int koff = half * 2;    // channel sub-offset inside a K=4 chunk

  const float* qb = qry + (size_t)b * (C_DIM * HW) + h * W_DIM;
  const float* kb = key + (size_t)b * (C_DIM * HW);
  float* ob = out + (size_t)b * ((size_t)HW * HW) + h * W_DIM; // + n*HW + w

  // Preload the whole query tile as B-fragments (K x N = 4 x 16 per chunk):
  // B[k][n] = qry[b][kc*4 + koff + {0,1}][h][w0 + n], lane n = l15.
  v2f qf[32];
#pragma unroll
  for (int kc = 0; kc < 32; ++kc) {
    const float* p = qb + (kc * 4 + koff) * HW + (w0 + l15);
    qf[kc].x = p[0];
    qf[kc].y = p[HW];
  }

  // Out-of-window (zero-padded) positions: each contributes exp(-10) at n=0.
  // Count analytically per query pixel; lanes 0..15 each own one pixel.
  // Only dh-group 0 performs this add (exactly once per pixel).
  if (blockIdx.y == 0) {
    int w = w0 + l15;
    int cnth = min(h + RAD, H_DIM - 1) - max(h - RAD, 0) + 1;
    int cntw = min(w + RAD, W_DIM - 1) - max(w - RAD, 0) + 1;
    int inval = PWIN * PWIN - cnth * cntw;
    if (half == 0 && inval > 0)
      atomicAdd(ob + w, (float)inval * EXP_M10);
  }

  const int wq = w0 + l15;  // this lane's query w (N index)
  const int dh0 = -RAD + (int)blockIdx.y * DH_PER_GROUP;

  for (int i = 0; i < DH_PER_GROUP; ++i) {    // wave-uniform
    int hh = h + dh0 + i;
    if (hh < 0 || hh >= H_DIM) continue;      // OOB rows handled analytically
    const float* krow = kb + hh * W_DIM;
    for (int t = 0; t < 3; ++t) {             // key column tiles w0-16, w0, w0+16
      int ww0 = w0 + (t - 1) * 16;
      int wwl = ww0 + l15;                    // A-fragment key pixel (M = l15)
      int wwc = min(max(wwl, 0), W_DIM - 1);  // clamp for address safety only
      const float* ka = krow + koff * HW + wwc;

      v8f acc = {};
#pragma unroll
      for (int kc = 0; kc < 32; ++kc) {
        // A (M x K = 16 x 4): A[m][k] = key[b][kc*4+koff+{0,1}][hh][ww0+m]
        v2f af;
        af.x = ka[0];
        af.y = ka[HW];
        ka += 4 * HW;
        acc = __builtin_amdgcn_wmma_f32_16x16x4_f32(
            /*neg_a=*/false, af, /*neg_b=*/false, qf[kc],
            /*c_mod=*/(short)0, acc, /*reuse_a=*/false, /*reuse_b=*/false);
      }

      // C/D layout: vgpr v, lane -> (M = v + half*8, N = l15).
      // Store: out[b][hh*64+ww][h][wq]; lanes 0..15 of a vgpr are 16
      // consecutive w addresses -> coalesced 64B segments, non-temporal.
#pragma unroll
      for (int v = 0; v < 8; ++v) {
        int m = v + half * 8;
        int ww = ww0 + m;
        int dw = ww - wq;
        if (ww >= 0 && ww < W_DIM && dw >= -RAD && dw <= RAD) {
          float c = acc[v];
          float val = (c == 0.0f) ? EXP_M10 : __expf(c * 10.0f);  // /TEMP = *10
          int n = hh * W_DIM + ww;
          float* dst = ob + (size_t)n * HW + wq;
          if (n == 0) atomicAdd(dst, val);  // n=0 collides with OOB mass
          else        __builtin_nontemporal_store(val, dst);
        }
      }
    }
  }
}

extern "C" void kernel_launch(void* const* d_in, const int* in_sizes, int n_in,
                              void* d_out, int out_size, void* d_ws, size_t ws_size,
                              hipStream_t stream) {
  (void)in_sizes; (void)n_in; (void)d_ws; (void)ws_size;
  const float* qry = (const float*)d_in[0];
  const float* key = (const float*)d_in[1];
  float* out = (float*)d_out;

  long long n4 = (long long)out_size / 4;   // out_size = 4*4096*64*64 fp32
  int zblocks = (int)((n4 + 255) / 256);
  zero_kernel<<<zblocks, 256, 0, stream>>>((v4f*)out, n4);

  corr_kernel<<<dim3(B_DIM * H_DIM, DH_GROUPS), 128, 0, stream>>>(qry, key, out);
}